// MyGATv2_70042326663943
// MI455X (gfx1250) — compile-verified
//
#include <hip/hip_runtime.h>

#define NN   50000
#define EE   800000
#define ETOT 850000   // EE + NN self-loops
#define GG   64
#define OC   10
#define HIDC 16

typedef float v2f __attribute__((ext_vector_type(2)));
typedef float v8f __attribute__((ext_vector_type(8)));

// ---------------------------------------------------------------------------
// WMMA fp32 GEMM: Y[M,NC] = X[M,K] @ W[K,NC] + bias.  One wave per 16x16 tile.
// Uses V_WMMA_F32_16X16X4_F32 (exact fp32 through the matrix pipe).
// A 16x4: lanes0-15 v0=K0,v1=K1 ; lanes16-31 v0=K2,v1=K3 (row = lane%16)
// B 4x16: lanes0-15 v0=K0row,v1=K1row ; lanes16-31 v0=K2,v1=K3 (col = lane%16)
// C 16x16: VGPR v -> row v (+8 for upper half-wave), col = lane%16
// ---------------------------------------------------------------------------
__global__ __launch_bounds__(32)
void wmma_gemm_bias(const float* __restrict__ X,
                    const float* __restrict__ W,
                    const float* __restrict__ bias,
                    float* __restrict__ Y,
                    int K, int NC) {
  const int lane = threadIdx.x & 31;
  const int m0   = blockIdx.x << 4;
  const int n0   = blockIdx.y << 4;
  const int half = lane >> 4;
  const int l16  = lane & 15;

  v8f c = {0.f, 0.f, 0.f, 0.f, 0.f, 0.f, 0.f, 0.f};
  const float* arow = X + (size_t)(m0 + l16) * K;
  const int col = n0 + l16;

  for (int k = 0; k < K; k += 4) {
    const int ka = k + (half << 1);
    v2f a;
    a.x = arow[ka];
    a.y = arow[ka + 1];
    const float* bp = W + (size_t)ka * NC + col;
    v2f b;
    b.x = bp[0];
    b.y = bp[NC];
    c = __builtin_amdgcn_wmma_f32_16x16x4_f32(false, a, false, b,
                                              (short)0, c, false, false);
  }

  const float bv = bias[col];
#pragma unroll
  for (int v = 0; v < 8; ++v) {
    Y[(size_t)(m0 + v + (half << 3)) * NC + col] = c[v] + bv;
  }
}

// ---------------------------------------------------------------------------
__global__ void fill_bits(float* __restrict__ p, long long n, unsigned bits) {
  long long i = (long long)blockIdx.x * blockDim.x + threadIdx.x;
  if (i < n) p[i] = __uint_as_float(bits);
}

__device__ __forceinline__ void atomicMaxF(float* addr, float v) {
  if (v >= 0.f) atomicMax((int*)addr, __float_as_int(v));
  else          atomicMin((unsigned int*)addr, __float_as_uint(v));
}

// pass 1: logits[e,h] = att_h . leaky_relu(xl[src] + xr[dst]) ; atomic max per dst
template <int H, int C>
__global__ __launch_bounds__(256)
void edge_logits_max(const int* __restrict__ ei,
                     const float* __restrict__ xl,
                     const float* __restrict__ xr,
                     const float* __restrict__ att,
                     float* __restrict__ logits,
                     float* __restrict__ mbuf) {
  constexpr int F = H * C;
  long long tid = (long long)blockIdx.x * blockDim.x + threadIdx.x;
  if (tid >= (long long)ETOT * H) return;
  int e = (int)(tid / H);
  int h = (int)(tid % H);
  int s = (e < EE) ? ei[e]      : (e - EE);
  int d = (e < EE) ? ei[EE + e] : (e - EE);
  const float4* pl = (const float4*)(xl + (size_t)s * F + h * C);
  const float4* pr = (const float4*)(xr + (size_t)d * F + h * C);
  const float*  pa = att + h * C;
  float acc = 0.f;
#pragma unroll
  for (int q = 0; q < C / 4; ++q) {
    float4 a = pl[q];
    float4 b = pr[q];
    float v0 = a.x + b.x, v1 = a.y + b.y, v2 = a.z + b.z, v3 = a.w + b.w;
    v0 = (v0 > 0.f) ? v0 : 0.2f * v0;   // leaky_relu, slope 0.2
    v1 = (v1 > 0.f) ? v1 : 0.2f * v1;
    v2 = (v2 > 0.f) ? v2 : 0.2f * v2;
    v3 = (v3 > 0.f) ? v3 : 0.2f * v3;
    acc += v0 * pa[q * 4 + 0] + v1 * pa[q * 4 + 1]
         + v2 * pa[q * 4 + 2] + v3 * pa[q * 4 + 3];
  }
  logits[tid] = acc;
  atomicMaxF(&mbuf[(size_t)d * H + h], acc);
}

// pass 2: ex = exp(logit - max[dst]); overwrite logits with ex; denom += ex
template <int H>
__global__ __launch_bounds__(256)
void edge_exp_denom(const int* __restrict__ ei,
                    float* __restrict__ logits,
                    const float* __restrict__ mbuf,
                    float* __restrict__ dbuf) {
  long long tid = (long long)blockIdx.x * blockDim.x + threadIdx.x;
  if (tid >= (long long)ETOT * H) return;
  int e = (int)(tid / H);
  int h = (int)(tid % H);
  int d = (e < EE) ? ei[EE + e] : (e - EE);
  float ex = __expf(logits[tid] - mbuf[(size_t)d * H + h]);
  logits[tid] = ex;
  atomicAdd(&dbuf[(size_t)d * H + h], ex);
}

// pass 3: out[dst, f] += (ex / denom[dst]) * xl[src, f]  (4 channels / thread)
template <int H, int C>
__global__ __launch_bounds__(256)
void edge_aggregate(const int* __restrict__ ei,
                    const float* __restrict__ logits,
                    const float* __restrict__ dbuf,
                    const float* __restrict__ xl,
                    float* __restrict__ out) {
  constexpr int F = H * C;
  constexpr int Q = F / 4;           // float4 groups per edge
  long long tid = (long long)blockIdx.x * blockDim.x + threadIdx.x;
  if (tid >= (long long)ETOT * Q) return;
  int e  = (int)(tid / Q);
  int q  = (int)(tid % Q);
  int f0 = q * 4;
  int h  = f0 / C;
  int s = (e < EE) ? ei[e]      : (e - EE);
  int d = (e < EE) ? ei[EE + e] : (e - EE);
  float alpha = logits[(size_t)e * H + h] / dbuf[(size_t)d * H + h];
  float4 v = *(const float4*)(xl + (size_t)s * F + f0);
  float* po = out + (size_t)d * F + f0;
  atomicAdd(po + 0, alpha * v.x);
  atomicAdd(po + 1, alpha * v.y);
  atomicAdd(po + 2, alpha * v.z);
  atomicAdd(po + 3, alpha * v.w);
}

// out = elu(out + bias)
__global__ __launch_bounds__(256)
void bias_elu(float* __restrict__ h, const float* __restrict__ bias, int F) {
  long long tid = (long long)blockIdx.x * blockDim.x + threadIdx.x;
  if (tid >= (long long)NN * F) return;
  int f = (int)(tid % F);
  float v = h[tid] + bias[f];
  h[tid] = (v > 0.f) ? v : (__expf(v) - 1.f);
}

// global mean pool (accumulate)
__global__ __launch_bounds__(256)
void pool_accum(const int* __restrict__ batch,
                const float* __restrict__ h,
                float* __restrict__ pooled,
                float* __restrict__ cnt) {
  long long tid = (long long)blockIdx.x * blockDim.x + threadIdx.x;
  if (tid >= (long long)NN * HIDC) return;
  int node = (int)(tid / HIDC);
  int f    = (int)(tid % HIDC);
  int g = batch[node];
  atomicAdd(&pooled[(size_t)g * HIDC + f], h[tid]);
  if (f == 0) atomicAdd(&cnt[g], 1.0f);
}

// final: out[g,o] = (pooled[g]/cnt[g]) @ fc_w + fc_b
__global__ __launch_bounds__(256)
void fc_out(const float* __restrict__ pooled,
            const float* __restrict__ cnt,
            const float* __restrict__ fcw,
            const float* __restrict__ fcb,
            float* __restrict__ out) {
  int tid = blockIdx.x * blockDim.x + threadIdx.x;
  if (tid >= GG * OC) return;
  int g = tid / OC, o = tid % OC;
  float inv = 1.0f / fmaxf(cnt[g], 1.0f);
  float acc = fcb[o];
#pragma unroll
  for (int f = 0; f < HIDC; ++f)
    acc += pooled[g * HIDC + f] * inv * fcw[f * OC + o];
  out[tid] = acc;
}

// ---------------------------------------------------------------------------
extern "C" void kernel_launch(void* const* d_in, const int* in_sizes, int n_in,
                              void* d_out, int out_size, void* d_ws, size_t ws_size,
                              hipStream_t stream) {
  const float* x     = (const float*)d_in[0];
  const int*   ei    = (const int*)d_in[1];
  const int*   batch = (const int*)d_in[2];

  // Resolve params layout: insertion order (Wl,bl,Wr,br,att,bias per layer,
  // then fc_w, fc_b) vs fully key-sorted flattening (fc_b, fc_w, then per
  // layer Wl,Wr,att,bias,bl,br). Distinguish via in_sizes[3].
  struct Layer { const float *Wl, *bl, *Wr, *br, *att, *bias; };
  Layer L[3];
  const float *fcw, *fcb;
  auto fp = [&](int i) { return (const float*)d_in[i]; };
  if (n_in > 3 && in_sizes[3] == 10) {       // sorted-key flattening
    fcb = fp(3); fcw = fp(4);
    int base = 5;
    for (int l = 0; l < 3; ++l, base += 6) {
      L[l].Wl = fp(base + 0); L[l].Wr   = fp(base + 1);
      L[l].att = fp(base + 2); L[l].bias = fp(base + 3);
      L[l].bl = fp(base + 4); L[l].br   = fp(base + 5);
    }
  } else {                                    // insertion-order flattening
    int base = 3;
    for (int l = 0; l < 3; ++l, base += 6) {
      L[l].Wl = fp(base + 0); L[l].bl   = fp(base + 1);
      L[l].Wr = fp(base + 2); L[l].br   = fp(base + 3);
      L[l].att = fp(base + 4); L[l].bias = fp(base + 5);
    }
    fcw = fp(21); fcb = fp(22);
  }

  // Workspace carve-up (floats); all offsets are multiples of 16 floats.
  float* ws = (float*)d_ws;
  size_t off = 0;
  float* xl   = ws + off; off += (size_t)NN * 128;
  float* xr   = ws + off; off += (size_t)NN * 128;
  float* hb   = ws + off; off += (size_t)NN * 128;   // layer-1/3 output
  float* lg   = ws + off; off += (size_t)ETOT * 8;   // per-edge logits / ex
  float* mb   = ws + off; off += (size_t)NN * 8;     // segment max
  float* db   = ws + off; off += (size_t)NN * 8;     // segment denom
  float* h2   = ws + off; off += (size_t)NN * HIDC;  // layer-2 output
  float* pooled = ws + off; off += GG * HIDC;
  float* cnt    = ws + off; off += GG;

  auto blocks = [](long long n) { return (unsigned)((n + 255) / 256); };

#define RUN_LAYER(INBUF, K, H, C, P, OUTBUF)                                   \
  do {                                                                         \
    constexpr int F_ = (H) * (C);                                              \
    dim3 gg(NN / 16, F_ / 16);                                                 \
    wmma_gemm_bias<<<gg, 32, 0, stream>>>(INBUF, (P).Wl, (P).bl, xl, K, F_);   \
    wmma_gemm_bias<<<gg, 32, 0, stream>>>(INBUF, (P).Wr, (P).br, xr, K, F_);   \
    long long nh = (long long)NN * (H);                                        \
    fill_bits<<<blocks(nh), 256, 0, stream>>>(mb, nh, 0xFF800000u);            \
    fill_bits<<<blocks(nh), 256, 0, stream>>>(db, nh, 0u);                     \
    long long nf = (long long)NN * F_;                                         \
    fill_bits<<<blocks(nf), 256, 0, stream>>>(OUTBUF, nf, 0u);                 \
    long long t1 = (long long)ETOT * (H);                                      \
    edge_logits_max<H, C><<<blocks(t1), 256, 0, stream>>>(ei, xl, xr,          \
                                                          (P).att, lg, mb);    \
    edge_exp_denom<H><<<blocks(t1), 256, 0, stream>>>(ei, lg, mb, db);         \
    long long t3 = (long long)ETOT * (F_ / 4);                                 \
    edge_aggregate<H, C><<<blocks(t3), 256, 0, stream>>>(ei, lg, db, xl,       \
                                                         OUTBUF);              \
    bias_elu<<<blocks(nf), 256, 0, stream>>>(OUTBUF, (P).bias, F_);            \
  } while (0)

  RUN_LAYER(x,  128, 8, HIDC, L[0], hb);   // conv1: 8 heads, concat -> [N,128]
  RUN_LAYER(hb, 128, 1, HIDC, L[1], h2);   // conv2: [N,16]
  RUN_LAYER(h2, HIDC, 1, HIDC, L[2], hb);  // conv3: [N,16] (reuse hb)
#undef RUN_LAYER

  fill_bits<<<1, 256, 0, stream>>>(pooled, GG * HIDC + GG, 0u);  // pooled+cnt
  pool_accum<<<blocks((long long)NN * HIDC), 256, 0, stream>>>(batch, hb,
                                                               pooled, cnt);
  fc_out<<<blocks(GG * OC), 256, 0, stream>>>(pooled, cnt, fcw, fcb,
                                              (float*)d_out);
}